// StructuredAttention_22703197127237
// MI455X (gfx1250) — compile-verified
//
#include <hip/hip_runtime.h>
#include <hip/hip_bf16.h>

// Structured attention (matrix-tree) pipeline for MI455X / gfx1250.
// fp32 end-to-end; all GEMMs via V_WMMA_F32_16X16X4_F32 (wave32 tiles).

namespace {

constexpr int kB    = 32;
constexpr int kT    = 1024;
constexpr int kSem  = 256;
constexpr int kStr  = 256;
constexpr int kRows = kB * kT;   // 32768
constexpr int kTp1  = kT + 1;    // 1025

typedef __attribute__((ext_vector_type(2))) float v2f;
typedef __attribute__((ext_vector_type(8))) float v8f;

__device__ __forceinline__ v8f wmma4(v2f a, v2f b, v8f c) {
  // D = A(16x4) * B(4x16) + C(16x16), fp32
  return __builtin_amdgcn_wmma_f32_16x16x4_f32(
      /*neg_a=*/false, a, /*neg_b=*/false, b,
      /*c_mod=*/(short)0, c, /*reuse_a=*/false, /*reuse_b=*/false);
}

// A-frag (16x4, MxK) from row-major [.. x ld], tile origin (m0,k0).
// lanes 0-15: M=lane, K=k0..k0+1 ; lanes 16-31: M=lane-16, K=k0+2..k0+3
__device__ __forceinline__ v2f frag_a_rm(const float* p, int ld, int m0, int k0, int lane) {
  int m  = m0 + (lane & 15);
  int kb = k0 + ((lane >> 4) << 1);
  v2f a;
  a.x = p[(size_t)m * ld + kb];
  a.y = p[(size_t)m * ld + kb + 1];
  return a;
}

// B-frag (4x16, KxN) from row-major K x N matrix (NN GEMM).
__device__ __forceinline__ v2f frag_b_nn(const float* p, int ld, int k0, int n0, int lane) {
  int n  = n0 + (lane & 15);
  int kb = k0 + ((lane >> 4) << 1);
  v2f b;
  b.x = p[(size_t)kb * ld + n];
  b.y = p[(size_t)(kb + 1) * ld + n];
  return b;
}

// B-frag for X @ W^T: W row-major [N x K]; B[k,n] = W[n,k].
__device__ __forceinline__ v2f frag_b_nt(const float* w, int ld, int k0, int n0, int lane) {
  int n  = n0 + (lane & 15);
  int kb = k0 + ((lane >> 4) << 1);
  v2f b;
  b.x = w[(size_t)n * ld + kb];
  b.y = w[(size_t)n * ld + kb + 1];
  return b;
}

// ---------------------------------------------------------------- split
// input [B,T,512] -> sem_v [B,T,256], str_v [B,T,256]
__global__ void k_split(const float* __restrict__ inp,
                        float* __restrict__ sem_v, float* __restrict__ str_v) {
  int i = blockIdx.x * blockDim.x + threadIdx.x;   // over kRows*256
  int bt = i >> 8, c = i & 255;
  const float* row = inp + (size_t)bt * 512;
  int dir = c >> 7, off = c & 127;
  sem_v[i] = row[dir * 256 + off];
  str_v[i] = row[dir * 256 + 128 + off];
}

// --------------------------------------------- tanh(X @ W^T + b), K=256
__global__ void k_gemm_nt_bias_tanh(const float* __restrict__ X, const float* __restrict__ W,
                                    const float* __restrict__ bias, float* __restrict__ out) {
  int wave = threadIdx.x >> 5, lane = threadIdx.x & 31;
  int tile = blockIdx.x * 8 + wave;      // tiles: (kRows/16) x 16, row-major
  int m0 = (tile >> 4) * 16, n0 = (tile & 15) * 16;
  v8f c = {};
#pragma unroll 4
  for (int k0 = 0; k0 < kStr; k0 += 4) {
    v2f a = frag_a_rm(X, kStr, m0, k0, lane);
    v2f b = frag_b_nt(W, kStr, k0, n0, lane);
    c = wmma4(a, b, c);
  }
  float bc = bias[n0 + (lane & 15)];
  int n = n0 + (lane & 15), mb = m0 + ((lane >> 4) << 3);
#pragma unroll
  for (int r = 0; r < 8; ++r) out[(size_t)(mb + r) * kStr + n] = tanhf(c[r] + bc);
}

// --------------------------------------------------------- U = tp @ Wbil
__global__ void k_gemm_nn(const float* __restrict__ X, const float* __restrict__ W,
                          float* __restrict__ out) {
  int wave = threadIdx.x >> 5, lane = threadIdx.x & 31;
  int tile = blockIdx.x * 8 + wave;
  int m0 = (tile >> 4) * 16, n0 = (tile & 15) * 16;
  v8f c = {};
#pragma unroll 4
  for (int k0 = 0; k0 < kStr; k0 += 4) {
    v2f a = frag_a_rm(X, kStr, m0, k0, lane);
    v2f b = frag_b_nn(W, kStr, k0, n0, lane);
    c = wmma4(a, b, c);
  }
  int n = n0 + (lane & 15), mb = m0 + ((lane >> 4) << 3);
#pragma unroll
  for (int r = 0; r < 8; ++r) out[(size_t)(mb + r) * kStr + n] = c[r];
}

// ------------------------------------------------ f_i = exp(str_v . wf)
__global__ void k_root_scores(const float* __restrict__ str_v, const float* __restrict__ wf,
                              float* __restrict__ fi) {
  int row = blockIdx.x * 8 + (threadIdx.x >> 5);
  int lane = threadIdx.x & 31;
  const float* x = str_v + (size_t)row * kStr;
  float s = 0.f;
  for (int k = lane; k < kStr; k += 32) s += x[k] * wf[k];
#pragma unroll
  for (int off = 16; off; off >>= 1) s += __shfl_xor(s, off, 32);
  if (lane == 0) fi[row] = expf(s);
}

// -------------------------- A[b] = exp(U[b] @ tc[b]^T) with zero diagonal
__global__ void k_scores(const float* __restrict__ U, const float* __restrict__ tc,
                         float* __restrict__ A) {
  int b = blockIdx.y;
  int wave = threadIdx.x >> 5, lane = threadIdx.x & 31;
  int tile = blockIdx.x * 8 + wave;     // 64 x 64 tiles
  int m0 = (tile >> 6) * 16, n0 = (tile & 63) * 16;
  const float* Ub  = U  + (size_t)b * kT * kStr;
  const float* tcb = tc + (size_t)b * kT * kStr;
  v8f c = {};
#pragma unroll 4
  for (int k0 = 0; k0 < kStr; k0 += 4) {
    v2f a = frag_a_rm(Ub, kStr, m0, k0, lane);
    v2f bf = frag_b_nt(tcb, kStr, k0, n0, lane);
    c = wmma4(a, bf, c);
  }
  float* Ab = A + (size_t)b * kT * kT;
  int n = n0 + (lane & 15), mb = m0 + ((lane >> 4) << 3);
#pragma unroll
  for (int r = 0; r < 8; ++r) {
    int i = mb + r;
    float v = expf(c[r]);
    if (i == n) v = 0.f;
    Ab[(size_t)i * kT + n] = v;
  }
}

// ------------------------------------------------ colsum[b,j] = sum_i A
__global__ void k_colsum(const float* __restrict__ A, float* __restrict__ cs) {
  int idx = blockIdx.x * blockDim.x + threadIdx.x;  // kRows
  int b = idx >> 10, j = idx & 1023;
  const float* Ab = A + (size_t)b * kT * kT;
  float s = 0.f;
  for (int i = 0; i < kT; ++i) s += Ab[(size_t)i * kT + j];
  cs[idx] = s;
}

// ------------------ L_bar: row0 = f_i, else -A + diag(colsum)
__global__ void k_build_L(const float* __restrict__ A, const float* __restrict__ cs,
                          const float* __restrict__ fi, float* __restrict__ L) {
  int idx = blockIdx.x * blockDim.x + threadIdx.x;  // B*T*T
  int j = idx & 1023, i = (idx >> 10) & 1023, b = idx >> 20;
  float v;
  if (i == 0) {
    v = fi[(b << 10) + j];
  } else {
    v = -A[(size_t)idx];
    if (i == j) v += cs[(b << 10) + j];
  }
  L[(size_t)idx] = v;
}

// ---------------- in-place Gauss-Jordan inverse, one workgroup per batch
__global__ void k_gj_inverse(float* __restrict__ M) {
  __shared__ float colk[kT];
  float* Mb = M + (size_t)blockIdx.x * kT * kT;
  int j = threadIdx.x;
  for (int k = 0; k < kT; ++k) {
    float piv = Mb[(size_t)k * kT + k];
    float inv = 1.0f / piv;
    colk[j] = Mb[(size_t)j * kT + k];                 // old column k
    float newkj = (j == k) ? inv : Mb[(size_t)k * kT + j] * inv;
    __syncthreads();
    Mb[(size_t)k * kT + j] = newkj;                   // new pivot row
    for (int i = 0; i < kT; ++i) {
      if (i == k) continue;
      float f = colk[i];
      float v = (j == k) ? (-f * inv) : (Mb[(size_t)i * kT + j] - f * newkj);
      Mb[(size_t)i * kT + j] = v;
    }
    __syncthreads();
  }
}

// --------------- d0, df[:, :, 0], diag(LLinv)
__global__ void k_rootmarg(const float* __restrict__ fi, const float* __restrict__ Minv,
                           float* __restrict__ df, float* __restrict__ d0,
                           float* __restrict__ diag) {
  int idx = blockIdx.x * blockDim.x + threadIdx.x;   // kRows
  int b = idx >> 10, t = idx & 1023;
  const float* Mb = Minv + (size_t)b * kT * kT;
  float v = fi[idx] * Mb[(size_t)t * kT];
  d0[idx] = v;
  df[(size_t)b * kT * kTp1 + (size_t)t * kTp1] = v;
  diag[idx] = Mb[(size_t)t * kT + t];
}

// --------------- dx = mask(A*diag_j - A*Minv^T); df[b,j,i+1] = dx[b,i,j]
__global__ void k_edge_df(const float* __restrict__ A, const float* __restrict__ Minv,
                          const float* __restrict__ diag, float* __restrict__ df) {
  int b = blockIdx.z;
  int i0 = blockIdx.y * 32, j0 = blockIdx.x * 32;
  __shared__ float tM[32][33];
  __shared__ float tD[32][33];
  int tx = threadIdx.x, ty = threadIdx.y;
  const float* Ab = A + (size_t)b * kT * kT;
  const float* Mb = Minv + (size_t)b * kT * kT;
  tM[ty][tx] = Mb[(size_t)(j0 + ty) * kT + (i0 + tx)];  // Minv rows j, cols i
  __syncthreads();
  int i = i0 + ty, j = j0 + tx;
  float a = Ab[(size_t)i * kT + j];
  float t1 = (j != 0) ? a * diag[(b << 10) + j] : 0.f;
  float t2 = (i != 0) ? a * tM[tx][ty] : 0.f;           // Minv[b,j,i]
  tD[ty][tx] = t1 - t2;
  __syncthreads();
  // write df[b, j0+ty, (i0+tx)+1] = dx[i0+tx, j0+ty]
  df[(size_t)b * kT * kTp1 + (size_t)(j0 + ty) * kTp1 + (i0 + tx) + 1] = tD[tx][ty];
}

// ---------------- pi = df @ ssr  (k=0 term = d0 (x) exparam; rest from sem)
__global__ void k_pi(const float* __restrict__ df, const float* __restrict__ sem,
                     const float* __restrict__ d0, const float* __restrict__ expar,
                     float* __restrict__ pi) {
  int b = blockIdx.y;
  int wave = threadIdx.x >> 5, lane = threadIdx.x & 31;
  int tile = blockIdx.x * 8 + wave;     // 64 x 16 tiles
  int m0 = (tile >> 4) * 16, n0 = (tile & 15) * 16;
  const float* dfb = df + (size_t)b * kT * kTp1;
  const float* sb  = sem + (size_t)b * kT * kSem;
  v8f c = {};
#pragma unroll 4
  for (int k0 = 0; k0 < kT; k0 += 4) {
    int m = m0 + (lane & 15);
    int kb = k0 + ((lane >> 4) << 1);
    v2f a;                                       // A[i,k] = df[b,i,k+1]
    a.x = dfb[(size_t)m * kTp1 + kb + 1];
    a.y = dfb[(size_t)m * kTp1 + kb + 2];
    v2f bf = frag_b_nn(sb, kSem, k0, n0, lane);
    c = wmma4(a, bf, c);
  }
  int n = n0 + (lane & 15), mb = m0 + ((lane >> 4) << 3);
  float e = expar[n];
  float* pb = pi + (size_t)b * kT * kSem;
#pragma unroll
  for (int r = 0; r < 8; ++r)
    pb[(size_t)(mb + r) * kSem + n] = c[r] + d0[(b << 10) + mb + r] * e;
}

// ---------------- ci = dx @ sem ; dx[b,i,k] = df[b,k,i+1]
__global__ void k_ci(const float* __restrict__ df, const float* __restrict__ sem,
                     float* __restrict__ ci) {
  int b = blockIdx.y;
  int wave = threadIdx.x >> 5, lane = threadIdx.x & 31;
  int tile = blockIdx.x * 8 + wave;
  int m0 = (tile >> 4) * 16, n0 = (tile & 15) * 16;
  const float* dfb = df + (size_t)b * kT * kTp1;
  const float* sb  = sem + (size_t)b * kT * kSem;
  v8f c = {};
#pragma unroll 4
  for (int k0 = 0; k0 < kT; k0 += 4) {
    int m = m0 + (lane & 15);
    int kb = k0 + ((lane >> 4) << 1);
    v2f a;                                       // A[i,k] = df[b,k,i+1]
    a.x = dfb[(size_t)kb * kTp1 + m + 1];
    a.y = dfb[(size_t)(kb + 1) * kTp1 + m + 1];
    v2f bf = frag_b_nn(sb, kSem, k0, n0, lane);
    c = wmma4(a, bf, c);
  }
  int n = n0 + (lane & 15), mb = m0 + ((lane >> 4) << 3);
  float* cb = ci + (size_t)b * kT * kSem;
#pragma unroll
  for (int r = 0; r < 8; ++r) cb[(size_t)(mb + r) * kSem + n] = c[r];
}

// ------------- ri = relu([sem|pi|ci] @ Wfz^T + bfz), K = 3*256
__global__ void k_final(const float* __restrict__ sem, const float* __restrict__ pi,
                        const float* __restrict__ ci, const float* __restrict__ Wfz,
                        const float* __restrict__ bfz, float* __restrict__ out) {
  int wave = threadIdx.x >> 5, lane = threadIdx.x & 31;
  int tile = blockIdx.x * 8 + wave;     // (kRows/16) x 16 tiles
  int m0 = (tile >> 4) * 16, n0 = (tile & 15) * 16;
  const float* segs[3] = {sem, pi, ci};
  v8f c = {};
#pragma unroll
  for (int s = 0; s < 3; ++s) {
    const float* X = segs[s];
    int koff = s * kSem;
#pragma unroll 4
    for (int k0 = 0; k0 < kSem; k0 += 4) {
      v2f a = frag_a_rm(X, kSem, m0, k0, lane);
      int n = n0 + (lane & 15);
      int kb = k0 + ((lane >> 4) << 1);
      v2f b;                                     // B[k,n] = Wfz[n, koff+k]
      b.x = Wfz[(size_t)n * (3 * kSem) + koff + kb];
      b.y = Wfz[(size_t)n * (3 * kSem) + koff + kb + 1];
      c = wmma4(a, b, c);
    }
  }
  int n = n0 + (lane & 15), mb = m0 + ((lane >> 4) << 3);
  float bc = bfz[n];
#pragma unroll
  for (int r = 0; r < 8; ++r) {
    float v = c[r] + bc;
    out[(size_t)(mb + r) * kSem + n] = v > 0.f ? v : 0.f;
  }
}

} // namespace

extern "C" void kernel_launch(void* const* d_in, const int* in_sizes, int n_in,
                              void* d_out, int out_size, void* d_ws, size_t ws_size,
                              hipStream_t stream) {
  (void)in_sizes; (void)n_in; (void)out_size; (void)ws_size;
  const float* inp    = (const float*)d_in[0];
  const float* Wp     = (const float*)d_in[1];
  const float* bp     = (const float*)d_in[2];
  const float* Wc     = (const float*)d_in[3];
  const float* bc     = (const float*)d_in[4];
  const float* wf     = (const float*)d_in[5];
  const float* Wbil   = (const float*)d_in[6];
  const float* expar  = (const float*)d_in[7];
  const float* Wfz    = (const float*)d_in[8];
  const float* bfz    = (const float*)d_in[9];

  float* ws = (float*)d_ws;
  size_t o = 0;
  float* sem_v = ws + o; o += (size_t)kRows * kSem;
  float* str_v = ws + o; o += (size_t)kRows * kStr;
  float* tp    = ws + o; o += (size_t)kRows * kStr;
  float* tc    = ws + o; o += (size_t)kRows * kStr;
  float* U     = ws + o; o += (size_t)kRows * kStr;
  float* pi    = ws + o; o += (size_t)kRows * kSem;
  float* ci    = ws + o; o += (size_t)kRows * kSem;
  float* fi    = ws + o; o += (size_t)kRows;
  float* cs    = ws + o; o += (size_t)kRows;
  float* diag  = ws + o; o += (size_t)kRows;
  float* d0    = ws + o; o += (size_t)kRows;
  float* A     = ws + o; o += (size_t)kB * kT * kT;
  float* L     = ws + o; o += (size_t)kB * kT * kT;   // becomes LLinv in place

  float* ri_out = (float*)d_out;
  float* df_out = ri_out + (size_t)kRows * kSem;      // [B, T, T+1]

  // 1. split into semantic / structural halves
  k_split<<<kRows * 256 / 256, 256, 0, stream>>>(inp, sem_v, str_v);
  // 2. tp / tc (tanh GEMMs)
  k_gemm_nt_bias_tanh<<<(kRows / 16) * 16 / 8, 256, 0, stream>>>(str_v, Wp, bp, tp);
  k_gemm_nt_bias_tanh<<<(kRows / 16) * 16 / 8, 256, 0, stream>>>(str_v, Wc, bc, tc);
  // 3. root scores
  k_root_scores<<<kRows / 8, 256, 0, stream>>>(str_v, wf, fi);
  // 4. U = tp @ Wbil
  k_gemm_nn<<<(kRows / 16) * 16 / 8, 256, 0, stream>>>(tp, Wbil, U);
  // 5. A = exp(U @ tc^T), zero diagonal
  k_scores<<<dim3(64 * 64 / 8, kB), 256, 0, stream>>>(U, tc, A);
  // 6. column sums
  k_colsum<<<kRows / 256, 256, 0, stream>>>(A, cs);
  // 7. Laplacian with root row
  k_build_L<<<(size_t)kB * kT * kT / 256, 256, 0, stream>>>(A, cs, fi, L);
  // 8. batched inverse (in place), one workgroup per batch, L2-resident
  k_gj_inverse<<<kB, kT, 0, stream>>>(L);
  // 9. root marginals + diag
  k_rootmarg<<<kRows / 256, 256, 0, stream>>>(fi, L, df_out, d0, diag);
  // 10. edge marginals -> df (transposed write via LDS tiles)
  k_edge_df<<<dim3(kT / 32, kT / 32, kB), dim3(32, 32), 0, stream>>>(A, L, diag, df_out);
  // 11/12. parent & child contexts
  k_pi<<<dim3(64 * 16 / 8, kB), 256, 0, stream>>>(df_out, sem_v, d0, expar, pi);
  k_ci<<<dim3(64 * 16 / 8, kB), 256, 0, stream>>>(df_out, sem_v, ci);
  // 13. fused output GEMM + bias + relu
  k_final<<<(kRows / 16) * 16 / 8, 256, 0, stream>>>(sem_v, pi, ci, Wfz, bfz, ri_out);
}